// GNNClassifier_GCN_33397665693792
// MI455X (gfx1250) — compile-verified
//
#include <hip/hip_runtime.h>

typedef float v2f __attribute__((ext_vector_type(2)));
typedef float v8f __attribute__((ext_vector_type(8)));

#define NFEAT 128

// ---------------------------------------------------------------- norm setup
__global__ __launch_bounds__(256) void k_init_deg(float* deg, int n) {
  int i = blockIdx.x * 256 + threadIdx.x;
  if (i < n) deg[i] = 1.0f;                       // self-loop weight
}

__global__ __launch_bounds__(256) void k_deg_scatter(const long long* __restrict__ dst,
                                                     const float* __restrict__ w,
                                                     float* deg, int e) {
  int i = blockIdx.x * 256 + threadIdx.x;
  if (i < e) atomicAdd(&deg[(int)dst[i]], w[i]);
}

__global__ __launch_bounds__(256) void k_dinv(float* deg, int n) {
  int i = blockIdx.x * 256 + threadIdx.x;
  if (i < n) {
    float d = deg[i];
    deg[i] = d > 0.0f ? rsqrtf(fmaxf(d, 1e-12f)) : 0.0f;
  }
}

// ---------------------------------------------------------------- WMMA GEMM
// Y[rows x 128] = X[rows x 128] @ W[128 x 128], fp32 via V_WMMA_F32_16X16X4_F32.
// 256 threads = 8 waves; wave w owns output column tile w; block owns 16 rows.
// A frag (16x4 f32): lanes 0-15 = M rows, lane-half selects K pair; 2 VGPRs.
// B frag (4x16 f32): lanes = N cols, lane-half selects K pair; 2 VGPRs.
// D (16x16 f32): VGPR v = row v (lanes 0-15) / row v+8 (lanes 16-31).
__global__ __launch_bounds__(256) void k_gemm128(const float* __restrict__ X,
                                                 const float* __restrict__ W,
                                                 float* __restrict__ Y) {
  const int lane = threadIdx.x & 31;
  const int wv   = threadIdx.x >> 5;         // 0..7 column tile
  const int m    = lane & 15;
  const int kh   = lane >> 4;                // 0 -> K+{0,1}, 1 -> K+{2,3}
  const size_t row0 = (size_t)blockIdx.x * 16;
  const float* xrow = X + (row0 + m) * NFEAT;
  const int ncol = wv * 16 + m;
  v8f acc = {};
  #pragma unroll 4
  for (int k = 0; k < NFEAT; k += 4) {
    const int kk = k + 2 * kh;
    v2f a; a.x = xrow[kk];                  a.y = xrow[kk + 1];                 // b64 load
    v2f b; b.x = W[(size_t)kk * NFEAT + ncol]; b.y = W[(size_t)(kk + 1) * NFEAT + ncol];
    acc = __builtin_amdgcn_wmma_f32_16x16x4_f32(false, a, false, b,
                                                (short)0, acc, false, false);
  }
  float* y = Y + row0 * NFEAT + ncol;
  const int mb = kh * 8;
  #pragma unroll
  for (int v = 0; v < 8; ++v) y[(size_t)(mb + v) * NFEAT] = acc[v];
}

// ---------------------------------------------------------------- scatter agg
__global__ __launch_bounds__(256) void k_zero(float* p, size_t n) {
  size_t i = (size_t)blockIdx.x * 256 + threadIdx.x;
  if (i < n) p[i] = 0.0f;
}

// one wave32 per edge; lane l covers features 4l..4l+3 (32*4 = 128)
__global__ __launch_bounds__(256) void k_edge_scatter(const float* __restrict__ H,
                                                      const long long* __restrict__ src,
                                                      const long long* __restrict__ dst,
                                                      const float* __restrict__ w,
                                                      const float* __restrict__ dinv,
                                                      float* agg, int nedges) {
  const int e = blockIdx.x * 8 + (threadIdx.x >> 5);
  if (e >= nedges) return;
  const int lane = threadIdx.x & 31;
  const int s = (int)src[e];
  const int d = (int)dst[e];
  const float nrm = dinv[s] * w[e] * dinv[d];
  const float4 hv = ((const float4*)(H + (size_t)s * NFEAT))[lane];
  float* a = agg + (size_t)d * NFEAT + lane * 4;
  atomicAdd(a + 0, hv.x * nrm);
  atomicAdd(a + 1, hv.y * nrm);
  atomicAdd(a + 2, hv.z * nrm);
  atomicAdd(a + 3, hv.w * nrm);
}

// agg += self-loop term (dinv[n]^2 * h_lin[n]) + bias, then ReLU (in place)
__global__ __launch_bounds__(256) void k_bias_relu_self(float* agg,
                                                        const float* __restrict__ hlin,
                                                        const float* __restrict__ dinv,
                                                        const float* __restrict__ bias,
                                                        size_t total) {
  size_t i = (size_t)blockIdx.x * 256 + threadIdx.x;
  if (i >= total) return;
  const int node = (int)(i >> 7);
  const int f = (int)(i & (NFEAT - 1));
  float sn = dinv[node]; sn *= sn;
  float v = agg[i] + hlin[i] * sn + bias[f];
  agg[i] = v > 0.0f ? v : 0.0f;
}

// ---------------------------------------------------------------- pool + FC
__global__ __launch_bounds__(256) void k_pool(const float* __restrict__ h,
                                              const long long* __restrict__ batch,
                                              float* pooled, float* cnt, int nnodes) {
  const int node = blockIdx.x * 8 + (threadIdx.x >> 5);
  if (node >= nnodes) return;
  const int lane = threadIdx.x & 31;
  const int g = (int)batch[node];
  const float4 hv = ((const float4*)(h + (size_t)node * NFEAT))[lane];
  float* p = pooled + (size_t)g * NFEAT + lane * 4;
  atomicAdd(p + 0, hv.x);
  atomicAdd(p + 1, hv.y);
  atomicAdd(p + 2, hv.z);
  atomicAdd(p + 3, hv.w);
  if (lane == 0) atomicAdd(&cnt[g], 1.0f);
}

__global__ __launch_bounds__(256) void k_fc(const float* __restrict__ pooled,
                                            const float* __restrict__ cnt,
                                            const float* __restrict__ Wfc,
                                            const float* __restrict__ bfc,
                                            float* __restrict__ out, int ngraphs) {
  const int i = blockIdx.x * 256 + threadIdx.x;   // i = g*16 + o
  if (i >= ngraphs * 16) return;
  const int g = i >> 4;
  const int o = i & 15;
  const float inv = 1.0f / fmaxf(cnt[g], 1.0f);
  const float* p = pooled + (size_t)g * NFEAT;
  float s = bfc[o];
  #pragma unroll 8
  for (int f = 0; f < NFEAT; ++f) s += p[f] * inv * Wfc[f * 16 + o];
  out[i] = s;
}

// ---------------------------------------------------------------- launcher
extern "C" void kernel_launch(void* const* d_in, const int* in_sizes, int n_in,
                              void* d_out, int out_size, void* d_ws, size_t ws_size,
                              hipStream_t stream) {
  const float*     x     = (const float*)d_in[0];
  const long long* eidx  = (const long long*)d_in[1];
  const float*     ew    = (const float*)d_in[2];
  const long long* batch = (const long long*)d_in[3];
  const float*     W1    = (const float*)d_in[4];
  const float*     b1    = (const float*)d_in[5];
  const float*     W2    = (const float*)d_in[6];
  const float*     b2    = (const float*)d_in[7];
  const float*     Wfc   = (const float*)d_in[8];
  const float*     bfc   = (const float*)d_in[9];
  float* out = (float*)d_out;

  const int N = in_sizes[0] / NFEAT;     // 50000
  const int E = in_sizes[2];             // 600000
  const int G = out_size / 16;           // 500
  const long long* src = eidx;           // edge_index[0]
  const long long* dst = eidx + E;       // edge_index[1]

  float* ws    = (float*)d_ws;
  float* bufA  = ws;                               // [N,128]  linear outputs
  float* bufB  = bufA + (size_t)N * NFEAT;         // [N,128]  aggregates / h
  float* dinv  = bufB + (size_t)N * NFEAT;         // [N]
  float* pooled = dinv + N;                        // [G,128]
  float* cnt   = pooled + (size_t)G * NFEAT;       // [G]

  const dim3 blk(256);
  const int gN  = (N + 255) / 256;
  const int gE  = (E + 255) / 256;
  const size_t nh = (size_t)N * NFEAT;
  const int gNH = (int)((nh + 255) / 256);
  const int gEw = (E + 7) / 8;                     // wave per edge
  const int gNw = (N + 7) / 8;                     // wave per node
  const int gMM = N / 16;                          // 3125 row tiles (N % 16 == 0)

  // D^{-1/2} with self loops
  k_init_deg<<<gN, blk, 0, stream>>>(dinv, N);
  k_deg_scatter<<<gE, blk, 0, stream>>>(dst, ew, dinv, E);
  k_dinv<<<gN, blk, 0, stream>>>(dinv, N);

  // layer 1: h1 = relu(Agg(x@W1) + b1)
  k_gemm128<<<gMM, blk, 0, stream>>>(x, W1, bufA);
  k_zero<<<gNH, blk, 0, stream>>>(bufB, nh);
  k_edge_scatter<<<gEw, blk, 0, stream>>>(bufA, src, dst, ew, dinv, bufB, E);
  k_bias_relu_self<<<gNH, blk, 0, stream>>>(bufB, bufA, dinv, b1, nh);

  // layer 2: h2 = relu(Agg(h1@W2) + b2)
  k_gemm128<<<gMM, blk, 0, stream>>>(bufB, W2, bufA);
  k_zero<<<gNH, blk, 0, stream>>>(bufB, nh);
  k_edge_scatter<<<gEw, blk, 0, stream>>>(bufA, src, dst, ew, dinv, bufB, E);
  k_bias_relu_self<<<gNH, blk, 0, stream>>>(bufB, bufA, dinv, b2, nh);

  // mean pool + FC head
  k_zero<<<(int)(((size_t)G * NFEAT + G + 255) / 256), blk, 0, stream>>>(pooled,
                                                       (size_t)G * NFEAT + G);
  k_pool<<<gNw, blk, 0, stream>>>(bufB, batch, pooled, cnt, N);
  k_fc<<<(G * 16 + 255) / 256, blk, 0, stream>>>(pooled, cnt, Wfc, bfc, out, G);
}